// PriceSeriesFeaturizer_70454643523997
// MI455X (gfx1250) — compile-verified
//
// PriceSeriesFeaturizer forward for MI455X (gfx1250), fp32 throughout.
// Core GEMM primitive: V_WMMA_F32_16X16X4_F32 (wave32, 16x16 tiles, K=4/instr).
// Working set ~61MB => L2-resident; compute bound => WMMA everywhere, fused
// GRU scans keep the recurrence on-WGP with LDS-resident hidden state.
// global_prefetch_b8 (via __builtin_prefetch) hides next-tile latency behind
// the WMMA inner loops.
//
// Assumptions (compile-only environment, documented):
//  * d_in flattening = JAX pytree order (sorted dict keys, tuples in order)
//    => 128 flat inputs, mapping computed below.
//  * ws_size >= ~61 MB (15,204,352 floats).

#include <hip/hip_runtime.h>
#include <hip/hip_bf16.h>
#include <math.h>

typedef float v2f __attribute__((ext_vector_type(2)));
typedef float v8f __attribute__((ext_vector_type(8)));

#if defined(__has_builtin)
#if __has_builtin(__builtin_amdgcn_wmma_f32_16x16x4_f32)
#define HAVE_WMMA_F32X4 1
#endif
#endif

__device__ __forceinline__ v8f wmma4(v2f a, v2f b, v8f c) {
#ifdef HAVE_WMMA_F32X4
  return __builtin_amdgcn_wmma_f32_16x16x4_f32(false, a, false, b, (short)0, c,
                                               false, false);
#else
  // compile-safe placeholder (never expected; would show wmma=0 in histogram)
  c[0] += a.x * b.x;
  return c;
#endif
}

__device__ __forceinline__ float wave_sum(float v) {
  for (int o = 16; o; o >>= 1) v += __shfl_xor(v, o, 32);
  return v;
}
__device__ __forceinline__ float wave_max(float v) {
  for (int o = 16; o; o >>= 1) v = fmaxf(v, __shfl_xor(v, o, 32));
  return v;
}

// ---------------------------------------------------------------------------
// Generic batched GEMM:  C = act(alpha * A@op(B) [+C_old] [+bias])
//  A: M x K, row-major, lda, batch stride sA
//  B: bT ? (N x K, C=A*B^T) : (K x N, C=A*B), ldb, batch stride sB
//  C: M x N, ldc, batch stride sC
//  shift/period: A row remap for conv taps: row r valid iff
//      0 <= (r % period)+shift < period, source row = r+shift. period<=0: off.
//  act: 0 none, 1 leaky(0.1), 3 tanh
// Workgroup: 256 thr (8 waves), tile 64Mx64N; wave tile 16Mx32N (2 wmma C).
// Next K-chunk is prefetched (global_prefetch_b8) while WMMAs run.
// ---------------------------------------------------------------------------
__launch_bounds__(256) __global__
void k_gemm(const float* __restrict__ A, int lda, long sA,
            const float* __restrict__ B, int ldb, long sB, int bT,
            float* __restrict__ C, int ldc, long sC,
            int M, int N, int Kd,
            const float* __restrict__ bias,
            float alpha, int acc, int act, int shift, int period) {
  __shared__ float As[64 * 32];
  __shared__ float Bs[64 * 32];
  const int tid = threadIdx.x;
  const int lane = tid & 31, wv = tid >> 5;
  const int mq = wv & 3, nq = wv >> 2;
  const int m0 = blockIdx.y * 64, n0 = blockIdx.x * 64;
  const long zA = (long)blockIdx.z * sA;
  const long zB = (long)blockIdx.z * sB;
  const long zC = (long)blockIdx.z * sC;
  const int ln = lane & 15;
  const int koff = (lane < 16) ? 0 : 2;

  v8f c0 = {0.f, 0.f, 0.f, 0.f, 0.f, 0.f, 0.f, 0.f};
  v8f c1 = c0;

  for (int kb = 0; kb < Kd; kb += 32) {
#pragma unroll
    for (int i = 0; i < 8; ++i) {
      int idx = tid + i * 256;
      int r = idx >> 5, cc = idx & 31;
      int kk = kb + cc;
      // A tile
      int g = m0 + r;
      float av = 0.f;
      if (g < M && kk < Kd) {
        if (period > 0) {
          int sp = (g % period) + shift;
          if ((unsigned)sp < (unsigned)period)
            av = A[zA + (long)(g + shift) * lda + kk];
        } else {
          av = A[zA + (long)g * lda + kk];
        }
      }
      As[r * 32 + cc] = av;
      // B tile, stored [n][k]
      int gn = n0 + r;
      float bv = 0.f;
      if (gn < N && kk < Kd)
        bv = bT ? B[zB + (long)gn * ldb + kk] : B[zB + (long)kk * ldb + gn];
      Bs[r * 32 + cc] = bv;
    }
    // prefetch next K-chunk (one line per tile row) while this chunk computes
    int kn = kb + 32;
    if (kn < Kd && tid < 64) {
      int g = m0 + tid;
      if (g < M) {
        if (period > 0) {
          int sp = (g % period) + shift;
          if ((unsigned)sp < (unsigned)period)
            __builtin_prefetch(&A[zA + (long)(g + shift) * lda + kn], 0, 0);
        } else {
          __builtin_prefetch(&A[zA + (long)g * lda + kn], 0, 0);
        }
      }
      int gn = n0 + tid;
      if (gn < N)
        __builtin_prefetch(bT ? &B[zB + (long)gn * ldb + kn]
                              : &B[zB + (long)kn * ldb + gn], 0, 0);
    }
    __syncthreads();
    const float* Ab  = &As[(mq * 16 + ln) * 32];
    const float* Bb0 = &Bs[(nq * 32 + ln) * 32];
    const float* Bb1 = &Bs[(nq * 32 + 16 + ln) * 32];
#pragma unroll
    for (int ks = 0; ks < 8; ++ks) {
      int k4 = ks * 4 + koff;
      v2f a  = {Ab[k4], Ab[k4 + 1]};
      v2f b0 = {Bb0[k4], Bb0[k4 + 1]};
      v2f b1 = {Bb1[k4], Bb1[k4 + 1]};
      c0 = wmma4(a, b0, c0);
      c1 = wmma4(a, b1, c1);
    }
    __syncthreads();
  }
#pragma unroll
  for (int r = 0; r < 8; ++r) {
    int row = m0 + mq * 16 + ((lane < 16) ? r : r + 8);
    int col0 = n0 + nq * 32 + ln;
    if (row < M && col0 < N) {
      float v = c0[r] * alpha;
      long off = zC + (long)row * ldc + col0;
      if (acc) v += C[off];
      if (bias) v += bias[col0];
      if (act == 1) v = (v >= 0.f) ? v : 0.1f * v;
      else if (act == 3) v = tanhf(v);
      C[off] = v;
    }
    int col1 = col0 + 16;
    if (row < M && col1 < N) {
      float v = c1[r] * alpha;
      long off = zC + (long)row * ldc + col1;
      if (acc) v += C[off];
      if (bias) v += bias[col1];
      if (act == 1) v = (v >= 0.f) ? v : 0.1f * v;
      else if (act == 3) v = tanhf(v);
      C[off] = v;
    }
  }
}

// ---------------------------------------------------------------------------
// Fused GRU scan: one WG = 16 sequences, hidden (16x128) in LDS.
// Per step: gh = h @ Whh^T via WMMA (24 N-tiles over 8 waves), then gates.
// gi: (Nseq, T, 384) with bih pre-added.  out cols [colOff,colOff+128).
// Next timestep's gi slab is prefetched during the gate phase.
// ---------------------------------------------------------------------------
__launch_bounds__(256) __global__
void k_gru_scan(const float* __restrict__ gi, const float* __restrict__ whh,
                const float* __restrict__ bhh, const float* __restrict__ h0,
                float* __restrict__ out, int T, int ldout, int colOff,
                int rev) {
  __shared__ float hs[16 * 128];
  __shared__ float ghs[16 * 384];
  const int tid = threadIdx.x;
  const int lane = tid & 31, wv = tid >> 5;
  const int ln = lane & 15;
  const int koff = (lane < 16) ? 0 : 2;
  const int seq0 = blockIdx.x * 16;

#pragma unroll
  for (int i = 0; i < 8; ++i) {
    int idx = tid + i * 256;
    int r = idx >> 7, cc = idx & 127;
    hs[idx] = h0 ? h0[(seq0 + r) * 128 + cc] : 0.f;
  }
  __syncthreads();

  for (int t = 0; t < T; ++t) {
    int ts = rev ? (T - 1 - t) : t;
    v8f cz = {0.f, 0.f, 0.f, 0.f, 0.f, 0.f, 0.f, 0.f};
    v8f c[3];
    c[0] = cz; c[1] = cz; c[2] = cz;
    for (int ks = 0; ks < 32; ++ks) {
      int k4 = ks * 4 + koff;
      v2f a = {hs[ln * 128 + k4], hs[ln * 128 + k4 + 1]};
#pragma unroll
      for (int j = 0; j < 3; ++j) {
        int n = (wv * 3 + j) * 16 + ln;
        v2f b = {whh[n * 128 + k4], whh[n * 128 + k4 + 1]};
        c[j] = wmma4(a, b, c[j]);
      }
    }
#pragma unroll
    for (int j = 0; j < 3; ++j) {
      int nb = (wv * 3 + j) * 16 + ln;
#pragma unroll
      for (int r = 0; r < 8; ++r) {
        int row = (lane < 16) ? r : r + 8;
        ghs[row * 384 + nb] = c[j][r];
      }
    }
    __syncthreads();
    // prefetch next timestep's gate slab while doing this step's pointwise
    if (t + 1 < T) {
      int ts2 = rev ? (T - 2 - t) : (t + 1);
      int r = tid >> 4, cc = (tid & 15) * 24;  // 256 thr cover 16x384 by lines
      __builtin_prefetch(&gi[((long)(seq0 + r) * T + ts2) * 384 + cc], 0, 0);
    }
#pragma unroll
    for (int i = 0; i < 8; ++i) {
      int idx = tid + i * 256;
      int r = idx >> 7, cc = idx & 127;
      long gb = ((long)(seq0 + r) * T + ts) * 384;
      float ir = gi[gb + cc];
      float iz = gi[gb + 128 + cc];
      float in_ = gi[gb + 256 + cc];
      float hr = ghs[r * 384 + cc] + bhh[cc];
      float hz = ghs[r * 384 + 128 + cc] + bhh[128 + cc];
      float hn = ghs[r * 384 + 256 + cc] + bhh[256 + cc];
      float rg = 1.f / (1.f + __expf(-(ir + hr)));
      float z = 1.f / (1.f + __expf(-(iz + hz)));
      float e2 = __expf(-2.f * (in_ + rg * hn));
      float n = (1.f - e2) / (1.f + e2);
      float hp = hs[idx];
      float h2 = (1.f - z) * n + z * hp;
      hs[idx] = h2;
      out[((long)(seq0 + r) * T + ts) * ldout + colOff + cc] = h2;
    }
    __syncthreads();
  }
}

// ---------------------------------------------------------------------------
// Small kernels
// ---------------------------------------------------------------------------
__global__ void k_embed_hist(const int* __restrict__ he,
                             const float* __restrict__ hf,
                             const float* __restrict__ em,
                             const float* __restrict__ ed,
                             const float* __restrict__ etd,
                             const float* __restrict__ etdl,
                             const float* __restrict__ W,
                             const float* __restrict__ bias,
                             float* __restrict__ X) {
  int i = blockIdx.x * blockDim.x + threadIdx.x;
  if (i >= 8192 * 256) return;
  int f = i & 255, row = i >> 8;
  const int* idx = he + row * 4;
  const float* fv = hf + row * 5;
  float s = bias[f];
  s += em[idx[0] * 2] * W[0 * 256 + f];
  s += em[idx[0] * 2 + 1] * W[1 * 256 + f];
  s += ed[idx[1] * 2] * W[2 * 256 + f];
  s += ed[idx[1] * 2 + 1] * W[3 * 256 + f];
  s += etd[idx[2]] * W[4 * 256 + f];
  s += etdl[idx[3]] * W[5 * 256 + f];
  s += fv[0] * W[6 * 256 + f];
  s += fv[1] * W[7 * 256 + f];
  s += fv[2] * W[8 * 256 + f];
  s += fv[3] * W[9 * 256 + f];
  s += fv[4] * W[10 * 256 + f];
  X[i] = s;
}

__global__ void k_embed_fut(const int* __restrict__ fe,
                            const float* __restrict__ em,
                            const float* __restrict__ ed,
                            const float* __restrict__ etd,
                            const float* __restrict__ etdl,
                            const float* __restrict__ W,
                            const float* __restrict__ bias,
                            const float* __restrict__ pos,
                            float* __restrict__ ff) {
  int i = blockIdx.x * blockDim.x + threadIdx.x;
  if (i >= 2048 * 256) return;
  int f = i & 255, row = i >> 8, q = row & 63;
  const int* idx = fe + row * 4;
  float s = bias[f] + pos[q * 256 + f];
  s += em[idx[0] * 2] * W[0 * 256 + f];
  s += em[idx[0] * 2 + 1] * W[1 * 256 + f];
  s += ed[idx[1] * 2] * W[2 * 256 + f];
  s += ed[idx[1] * 2 + 1] * W[3 * 256 + f];
  s += etd[idx[2]] * W[4 * 256 + f];
  s += etdl[idx[3]] * W[5 * 256 + f];
  ff[i] = s;
}

__global__ void k_repack_hfd(const float* __restrict__ w, float* __restrict__ o) {
  int i = blockIdx.x * blockDim.x + threadIdx.x;   // (H,F,D)->(F,H*D)
  if (i >= 65536) return;
  int f = i >> 8, n = i & 255, h = n >> 5, d = n & 31;
  o[i] = w[(h * 256 + f) * 32 + d];
}

__global__ void k_repack_conv(const float* __restrict__ w, float* __restrict__ o) {
  int i = blockIdx.x * blockDim.x + threadIdx.x;   // (O,I,5)->(5,I,O)
  if (i >= 327680) return;
  int k = i / 65536, r = i % 65536, fi = r >> 8, fo = r & 255;
  o[i] = w[(fo * 256 + fi) * 5 + k];
}

__global__ void k_softmax(float* __restrict__ x, int L, long nrows) {
  long w = ((long)blockIdx.x * blockDim.x + threadIdx.x) >> 5;
  int lane = threadIdx.x & 31;
  if (w >= nrows) return;
  float* p = x + w * (long)L;
  int cnt = L >> 5;
  float v[8];
  float m = -1e30f;
  for (int i = 0; i < cnt; ++i) {
    v[i] = p[lane + i * 32];
    m = fmaxf(m, v[i]);
  }
  m = wave_max(m);
  float s = 0.f;
  for (int i = 0; i < cnt; ++i) {
    v[i] = __expf(v[i] - m);
    s += v[i];
  }
  s = wave_sum(s);
  float inv = 1.f / s;
  for (int i = 0; i < cnt; ++i) p[lane + i * 32] = v[i] * inv;
}

// leaky(0.1) + LN over D=32 + head-concat residual add into y (B,S,256)
__global__ void k_ln_head_add(const float* __restrict__ ab,
                              const float* __restrict__ g,
                              const float* __restrict__ bb,
                              float* __restrict__ y) {
  int w = (blockIdx.x * blockDim.x + threadIdx.x) >> 5;  // (b*H+h)*S+s
  int lane = threadIdx.x & 31;
  float v = ab[(long)w * 32 + lane];
  v = (v >= 0.f) ? v : 0.1f * v;
  float m = wave_sum(v) * (1.f / 32.f);
  float d = v - m;
  float var = wave_sum(d * d) * (1.f / 32.f);
  int h = (w >> 8) & 7;
  int b = w >> 11;
  int s = w & 255;
  float o = d * rsqrtf(var + 1e-5f) * g[h * 32 + lane] + bb[h * 32 + lane];
  y[((long)(b * 256 + s)) * 256 + h * 32 + lane] += o;
}

__global__ void k_add(float* __restrict__ dst, const float* __restrict__ a,
                      const float* __restrict__ b, long n) {
  long i = (long)blockIdx.x * blockDim.x + threadIdx.x;
  if (i >= n) return;
  float v = dst[i] + a[i];
  if (b) v += b[i];
  dst[i] = v;
}

__global__ void k_addpos(const float* __restrict__ ff,
                         const float* __restrict__ pos,
                         float* __restrict__ xa, long n) {
  long i = (long)blockIdx.x * blockDim.x + threadIdx.x;
  if (i >= n) return;
  xa[i] = ff[i] + pos[i & 16383];
}

// out = ff + LN_256(leaky(af,0.2)) with fa_ln
__global__ void k_final(const float* __restrict__ ff,
                        const float* __restrict__ af,
                        const float* __restrict__ g,
                        const float* __restrict__ bb,
                        float* __restrict__ out) {
  int w = (blockIdx.x * blockDim.x + threadIdx.x) >> 5;  // row b*64+q
  int lane = threadIdx.x & 31;
  const float* ar = af + (long)w * 256;
  float v[8];
  float s = 0.f;
#pragma unroll
  for (int i = 0; i < 8; ++i) {
    float t = ar[lane + 32 * i];
    v[i] = (t >= 0.f) ? t : 0.2f * t;
    s += v[i];
  }
  float m = wave_sum(s) * (1.f / 256.f);
  float var = 0.f;
#pragma unroll
  for (int i = 0; i < 8; ++i) {
    float d = v[i] - m;
    var += d * d;
  }
  var = wave_sum(var) * (1.f / 256.f);
  float rs = rsqrtf(var + 1e-5f);
#pragma unroll
  for (int i = 0; i < 8; ++i) {
    int col = lane + 32 * i;
    out[(long)w * 256 + col] =
        ff[(long)w * 256 + col] + (v[i] - m) * rs * g[col] + bb[col];
  }
}

// ---------------------------------------------------------------------------
// Host orchestration
// ---------------------------------------------------------------------------
static inline void gemm(hipStream_t st, const float* A, int lda, long sA,
                        const float* B, int ldb, long sB, int bT, float* C,
                        int ldc, long sC, int M, int N, int Kd,
                        const float* bias, float alpha, int acc, int act,
                        int shift, int period, int batch) {
  dim3 g((unsigned)((N + 63) / 64), (unsigned)((M + 63) / 64),
         (unsigned)batch);
  k_gemm<<<g, 256, 0, st>>>(A, lda, sA, B, ldb, sB, bT, C, ldc, sC, M, N, Kd,
                            bias, alpha, acc, act, shift, period);
}

extern "C" void kernel_launch(void* const* d_in, const int* in_sizes, int n_in,
                              void* d_out, int out_size, void* d_ws,
                              size_t ws_size, hipStream_t stream) {
  (void)in_sizes; (void)n_in; (void)out_size; (void)ws_size;
  const int* h_emb = (const int*)d_in[0];
  const float* h_float = (const float*)d_in[1];
  const int* f_emb = (const int*)d_in[2];
  const float* W_h2h = (const float*)d_in[3];
  const float* W_in_fut = (const float*)d_in[4];
  const float* W_in_hist = (const float*)d_in[5];
  const float* Wq_fut = (const float*)d_in[6];
  const float* b_h2h = (const float*)d_in[7];
  const float* b_in_fut = (const float*)d_in[8];
  const float* b_in_hist = (const float*)d_in[9];
  const float* emb_dow = (const float*)d_in[102];
  const float* emb_month = (const float*)d_in[103];
  const float* emb_pos = (const float*)d_in[104];
  const float* emb_td = (const float*)d_in[105];
  const float* emb_tdl = (const float*)d_in[106];
  const float* fa_Wk = (const float*)d_in[107];
  const float* fa_Wq = (const float*)d_in[108];
  const float* fa_Wv = (const float*)d_in[109];
  const float* fa_ln_b = (const float*)d_in[110];
  const float* fa_ln_g = (const float*)d_in[111];

  float* ws = (float*)d_ws;
  const long TWO_M = 2097152;
  float* X = ws + 0 * TWO_M;     // h_feat
  float* Y = ws + 1 * TWO_M;     // block working / f_feat
  float* Q = ws + 2 * TWO_M;
  float* KB = ws + 3 * TWO_M;
  float* V = ws + 4 * TWO_M;
  float* SC = ws + 5 * TWO_M;
  float* AB = ws + 6 * TWO_M;
  float* PK = ws + 7 * TWO_M;    // weight repack region (524288 floats)
  float* PKQ = PK, *PKK = PK + 65536, *PKV = PK + 131072, *PKC = PK + 196608;
  // GRU reuse of attention regions:
  float* GI = Q;                 // 3,145,728 floats spills into KB region (dead)
  float* G1 = V;
  float* G2 = SC;

  const float rD = 0.17677669529663687f;  // 1/sqrt(32)
  const float rF = 0.0625f;               // 1/sqrt(256)

  // ---- history embedding + input projection -> X (B*S, 256)
  k_embed_hist<<<8192, 256, 0, stream>>>(h_emb, h_float, emb_month, emb_dow,
                                         emb_td, emb_tdl, W_in_hist, b_in_hist,
                                         X);

  // ---- encoder blocks
  for (int blk = 0; blk < 4; ++blk) {
    int pb = 10 + blk * 23;
    const float* Wk = (const float*)d_in[pb + 0];
    const float* Wq = (const float*)d_in[pb + 1];
    const float* Wv = (const float*)d_in[pb + 2];
    const float* conv_b = (const float*)d_in[pb + 3];
    const float* conv_w = (const float*)d_in[pb + 4];
    const float* ln_b = (const float*)d_in[pb + 21];
    const float* ln_g = (const float*)d_in[pb + 22];

    k_repack_hfd<<<256, 256, 0, stream>>>(Wq, PKQ);
    k_repack_hfd<<<256, 256, 0, stream>>>(Wk, PKK);
    k_repack_hfd<<<256, 256, 0, stream>>>(Wv, PKV);
    k_repack_conv<<<1280, 256, 0, stream>>>(conv_w, PKC);

    // conv (5 shifted GEMM taps) -> Y, leaky(0.1)+bias on last tap
    for (int k = 0; k < 5; ++k)
      gemm(stream, X, 256, 0, PKC + k * 65536, 256, 0, 0, Y, 256, 0, 8192, 256,
           256, (k == 4) ? conv_b : nullptr, 1.f, (k > 0), (k == 4) ? 1 : 0,
           k - 2, 256, 1);

    // QKV projections
    gemm(stream, Y, 256, 0, PKQ, 256, 0, 0, Q, 256, 0, 8192, 256, 256, nullptr,
         1.f, 0, 0, 0, 0, 1);
    gemm(stream, Y, 256, 0, PKK, 256, 0, 0, KB, 256, 0, 8192, 256, 256,
         nullptr, 1.f, 0, 0, 0, 0, 1);
    gemm(stream, Y, 256, 0, PKV, 256, 0, 0, V, 256, 0, 8192, 256, 256, nullptr,
         1.f, 0, 0, 0, 0, 1);

    // attention, per head (scores buffer B*S*S reused)
    for (int h = 0; h < 8; ++h) {
      gemm(stream, Q + h * 32, 256, 65536, KB + h * 32, 256, 65536, 1, SC, 256,
           65536, 256, 256, 32, nullptr, rD, 0, 0, 0, 0, 32);
      k_softmax<<<1024, 256, 0, stream>>>(SC, 256, 8192);
      gemm(stream, SC, 256, 65536, V + h * 32, 256, 65536, 0, AB + h * 8192,
           32, 65536, 256, 32, 256, nullptr, 1.f, 0, 0, 0, 0, 32);
    }
    k_ln_head_add<<<8192, 256, 0, stream>>>(AB, ln_g, ln_b, Y);

    // BiGRU (2 layers x {fwd,bwd}), seqs = 256, T = 32
    for (int layer = 0; layer < 2; ++layer) {
      const float* inp = (layer == 0) ? Y : G1;
      float* outp = (layer == 0) ? G1 : G2;
      for (int dir = 0; dir < 2; ++dir) {  // 0=fwd (leaves +4), 1=bwd (+0)
        int gb = pb + 5 + layer * 8 + ((dir == 0) ? 4 : 0);
        const float* Wih = (const float*)d_in[gb + 0];
        const float* Whh = (const float*)d_in[gb + 1];
        const float* bih = (const float*)d_in[gb + 2];
        const float* bhh = (const float*)d_in[gb + 3];
        gemm(stream, inp, 256, 0, Wih, 256, 0, 1, GI, 384, 0, 8192, 384, 256,
             bih, 1.f, 0, 0, 0, 0, 1);
        k_gru_scan<<<16, 256, 0, stream>>>(GI, Whh, bhh, nullptr, outp, 32,
                                           256, dir * 128, dir);
      }
    }
    k_add<<<8192, 256, 0, stream>>>(X, Y, G2, TWO_M);  // X += Y + gru_out
  }

  // ---- future head
  float* FF = Y;
  float* Q2 = Q;
  float* SCF = KB;
  float* T0 = AB;  // tanh(h_last @ W_h2h + b) => h0 (4,B,128) flat view
  float* GIF = Q;
  float* G1F = KB;
  float* G2F = V;
  float* XA = Q;
  float* FQ = KB;
  float* FK = V;
  float* FV = SC;
  float* WA = AB + 32768;
  float* AF = AB + 262144;

  k_embed_fut<<<2048, 256, 0, stream>>>(f_emb, emb_month, emb_dow, emb_td,
                                        emb_tdl, W_in_fut, b_in_fut, emb_pos,
                                        FF);
  // cross attention onto h_feat (X)
  gemm(stream, FF, 256, 0, Wq_fut, 256, 0, 0, Q2, 256, 0, 2048, 256, 256,
       nullptr, 1.f, 0, 0, 0, 0, 1);
  gemm(stream, Q2, 256, 16384, X, 256, 65536, 1, SCF, 256, 16384, 64, 256, 256,
       nullptr, rF, 0, 0, 0, 0, 32);
  k_softmax<<<1024, 256, 0, stream>>>(SCF, 64, 8192);  // chunks of 64 along s
  gemm(stream, SCF, 256, 16384, X, 256, 65536, 0, FF, 256, 16384, 64, 256, 256,
       nullptr, 1.f, 1, 0, 0, 0, 32);
  // h0 = tanh(h_feat[:,-1] @ W_h2h + b_h2h), flat (4,B,128)
  gemm(stream, X + 255 * 256, 65536, 0, W_h2h, 512, 0, 0, T0, 512, 0, 32, 512,
       256, b_h2h, 1.f, 0, 3, 0, 0, 1);
  // future BiGRU: seqs = 32, T = 64
  for (int layer = 0; layer < 2; ++layer) {
    const float* inp = (layer == 0) ? FF : G1F;
    float* outp = (layer == 0) ? G1F : G2F;
    for (int dir = 0; dir < 2; ++dir) {
      int gb = 112 + layer * 8 + ((dir == 0) ? 4 : 0);
      const float* Wih = (const float*)d_in[gb + 0];
      const float* Whh = (const float*)d_in[gb + 1];
      const float* bih = (const float*)d_in[gb + 2];
      const float* bhh = (const float*)d_in[gb + 3];
      gemm(stream, inp, 256, 0, Wih, 256, 0, 1, GIF, 384, 0, 2048, 384, 256,
           bih, 1.f, 0, 0, 0, 0, 1);
      const float* h0p = T0 + (2 * layer + dir) * 32 * 128;  // fwd=2l, bwd=2l+1
      k_gru_scan<<<2, 256, 0, stream>>>(GIF, Whh, bhh, h0p, outp, 64, 256,
                                        dir * 128, dir);
    }
  }
  k_add<<<2048, 256, 0, stream>>>(FF, G2F, nullptr, 524288);
  // final self-attention over future positions
  k_addpos<<<2048, 256, 0, stream>>>(FF, emb_pos, XA, 524288);
  gemm(stream, XA, 256, 0, fa_Wq, 256, 0, 0, FQ, 256, 0, 2048, 256, 256,
       nullptr, 1.f, 0, 0, 0, 0, 1);
  gemm(stream, XA, 256, 0, fa_Wk, 256, 0, 0, FK, 256, 0, 2048, 256, 256,
       nullptr, 1.f, 0, 0, 0, 0, 1);
  gemm(stream, XA, 256, 0, fa_Wv, 256, 0, 0, FV, 256, 0, 2048, 256, 256,
       nullptr, 1.f, 0, 0, 0, 0, 1);
  gemm(stream, FQ, 256, 16384, FK, 256, 16384, 1, WA, 64, 4096, 64, 64, 256,
       nullptr, rF, 0, 0, 0, 0, 32);
  k_softmax<<<256, 256, 0, stream>>>(WA, 64, 2048);
  gemm(stream, WA, 64, 4096, FV, 256, 16384, 0, AF, 256, 16384, 64, 256, 64,
       nullptr, 1.f, 0, 0, 0, 0, 32);
  k_final<<<256, 256, 0, stream>>>(FF, AF, fa_ln_g, fa_ln_b, (float*)d_out);
}